// GCN_2302102471490
// MI455X (gfx1250) — compile-verified
//
#include <hip/hip_runtime.h>
#include <hip/hip_bf16.h>

typedef __attribute__((ext_vector_type(2))) float v2f;
typedef __attribute__((ext_vector_type(8))) float v8f;

// ---------------- degree / normalization ----------------
__global__ void k_fill(float* __restrict__ p, float v, int n) {
  int i = blockIdx.x * blockDim.x + threadIdx.x;
  if (i < n) p[i] = v;
}

__global__ void k_deg_edges(const int* __restrict__ dst, float* __restrict__ deg, int E) {
  int i = blockIdx.x * blockDim.x + threadIdx.x;
  if (i < E) atomicAdd(deg + dst[i], 1.0f);
}

__global__ void k_dinv(float* __restrict__ deg, int n) {
  int i = blockIdx.x * blockDim.x + threadIdx.x;
  if (i < n) deg[i] = rsqrtf(fmaxf(deg[i], 1.0f));
}

// ---------------- WMMA f32 GEMM: out[N,M] = A[N,K] @ W[K,M] ----------------
// Block: 256 threads = 8 waves. Block computes 16 rows x 128 cols.
// Wave w computes the 16x16 tile at cols [blockIdx.y*128 + w*16).
// A-tile (16 x K) staged in LDS (row stride K+4 to spread banks: (K+4)%64==4).
__global__ void __launch_bounds__(256) k_gemm_wmma(
    const float* __restrict__ A, const float* __restrict__ W,
    float* __restrict__ out, int N, int K, int M) {
  __shared__ float As[16 * (256 + 4)];
  const int ldsK = K + 4;
  const int rowBase = blockIdx.x * 16;
  const int colBase = blockIdx.y * 128 + (threadIdx.x >> 5) * 16;

  // Cooperative load of the 16 x K A-tile (zero-pad rows >= N).
  for (int e = threadIdx.x; e < 16 * K; e += 256) {
    int r = e / K;
    int c = e - r * K;
    int g = rowBase + r;
    As[r * ldsK + c] = (g < N) ? A[(size_t)g * K + c] : 0.0f;
  }
  __syncthreads();

  const int lane = threadIdx.x & 31;
  const int half = lane >> 4;   // 0: lanes 0-15, 1: lanes 16-31
  const int m    = lane & 15;   // A row (M) / D col (N)
  const int col  = colBase + m;

  v8f acc = {};
  for (int k = 0; k < K; k += 4) {
    const int kk = k + 2 * half;     // ISA: VGPR0 holds K={0,2}, VGPR1 K={1,3} per lane-half
    v2f a, b;
    a.x = As[m * ldsK + kk];
    a.y = As[m * ldsK + kk + 1];
    b.x = W[(size_t)kk * M + col];
    b.y = W[(size_t)(kk + 1) * M + col];
    acc = __builtin_amdgcn_wmma_f32_16x16x4_f32(
        /*neg_a=*/false, a, /*neg_b=*/false, b,
        /*c_mod=*/(short)0, acc, /*reuse_a=*/false, /*reuse_b=*/false);
  }

  // D layout: VGPR r -> row (r + 8*half), lane%16 -> col
  for (int r = 0; r < 8; ++r) {
    int row = rowBase + r + 8 * half;
    if (row < N) out[(size_t)row * M + col] = acc[r];
  }
}

// ---------------- aggregation ----------------
// Self-loop contribution (no atomics needed): agg[i,c] = hW[i,c] * dinv[i]^2
__global__ void k_selfloop(const float* __restrict__ hW, const float* __restrict__ dinv,
                           float* __restrict__ agg, int N, int C) {
  long long i = (long long)blockIdx.x * blockDim.x + threadIdx.x;
  long long total = (long long)N * C;
  if (i < total) {
    int node = (int)(i / C);
    float di = dinv[node];
    agg[i] = hW[i] * di * di;
  }
}

// One wave per edge: agg[dst] += hW[src] * dinv[src]*dinv[dst]
__global__ void k_edge_scatter(const int* __restrict__ src, const int* __restrict__ dst,
                               const float* __restrict__ dinv, const float* __restrict__ hW,
                               float* __restrict__ agg, int E, int C) {
  int wave = (int)(((long long)blockIdx.x * blockDim.x + threadIdx.x) >> 5);
  int lane = threadIdx.x & 31;
  if (wave >= E) return;
  int s = src[wave];
  int d = dst[wave];
  float nrm = dinv[s] * dinv[d];
  const float4* hp = (const float4*)(hW + (size_t)s * C);
  float* ap = agg + (size_t)d * C;
  int nchunk = C >> 7;  // C / (32 lanes * 4 floats); C is 128 or 256
  for (int j = 0; j < nchunk; ++j) {
    int v4 = j * 32 + lane;
    float4 v = hp[v4];
    int base = v4 * 4;
    atomicAdd(ap + base + 0, v.x * nrm);
    atomicAdd(ap + base + 1, v.y * nrm);
    atomicAdd(ap + base + 2, v.z * nrm);
    atomicAdd(ap + base + 3, v.w * nrm);
  }
}

// ---------------- bias + relu + layernorm (wave per node, wave32) ----------------
__global__ void k_post(const float* __restrict__ agg, const float* __restrict__ bias,
                       float* __restrict__ out, int N, int C) {
  int node = (int)(((long long)blockIdx.x * blockDim.x + threadIdx.x) >> 5);
  int lane = threadIdx.x & 31;
  if (node >= N) return;
  const float* row = agg + (size_t)node * C;
  float v[8];                 // C <= 256 -> at most 8 channels per lane
  int nch = C >> 5;
  float s = 0.0f;
  for (int j = 0; j < nch; ++j) {
    float t = fmaxf(row[j * 32 + lane] + bias[j * 32 + lane], 0.0f);  // relu(conv + b)
    v[j] = t;
    s += t;
  }
  for (int off = 16; off > 0; off >>= 1) s += __shfl_xor(s, off, 32);
  float mean = s / (float)C;
  float q = 0.0f;
  for (int j = 0; j < nch; ++j) {
    float d = v[j] - mean;
    q += d * d;
  }
  for (int off = 16; off > 0; off >>= 1) q += __shfl_xor(q, off, 32);
  float rstd = rsqrtf(q / (float)C + 1e-5f);
  float* o = out + (size_t)node * C;
  for (int j = 0; j < nch; ++j) o[j * 32 + lane] = (v[j] - mean) * rstd;
}

// ---------------- pooling ----------------
__global__ void k_pool_accum(const float* __restrict__ h, const int* __restrict__ batch,
                             float* __restrict__ pool, int N, int C) {
  long long i = (long long)blockIdx.x * blockDim.x + threadIdx.x;
  long long total = (long long)N * C;
  if (i < total) {
    int node = (int)(i / C);
    int c = (int)(i - (long long)node * C);
    atomicAdd(pool + (size_t)batch[node] * C + c, h[i]);
  }
}

__global__ void k_count(const int* __restrict__ batch, float* __restrict__ cnt, int N) {
  int i = blockIdx.x * blockDim.x + threadIdx.x;
  if (i < N) atomicAdd(cnt + batch[i], 1.0f);
}

__global__ void k_pool_final(float* __restrict__ pool, const float* __restrict__ cnt,
                             int G, int C) {
  int i = blockIdx.x * blockDim.x + threadIdx.x;
  if (i < G * C) pool[i] /= fmaxf(cnt[i / C], 1.0f);
}

// ---------------- orchestration ----------------
static inline size_t align256(size_t x) { return (x + 255) & ~(size_t)255; }

extern "C" void kernel_launch(void* const* d_in, const int* in_sizes, int n_in,
                              void* d_out, int out_size, void* d_ws, size_t ws_size,
                              hipStream_t stream) {
  const float* x     = (const float*)d_in[0];
  const int*   ei    = (const int*)d_in[1];
  const int*   batch = (const int*)d_in[2];
  // d_in[3] = num_graphs (device scalar; G derived from out_size instead)
  const float* W0 = (const float*)d_in[4];
  const float* b0 = (const float*)d_in[5];
  const float* W1 = (const float*)d_in[6];
  const float* b1 = (const float*)d_in[7];
  const float* W2 = (const float*)d_in[8];
  const float* b2 = (const float*)d_in[9];

  const int N  = in_sizes[2];           // 50000 (batch length)
  const int E  = in_sizes[1] / 2;       // 600000
  const int IN = in_sizes[0] / N;       // 128
  const int H0 = in_sizes[5];           // 256
  const int H1 = in_sizes[7];           // 256
  const int H2 = in_sizes[9];           // 128
  const int G  = out_size / H2 - N;     // 64

  const int* srcp = ei;
  const int* dstp = ei + E;

  // Workspace partition
  char* ws = (char*)d_ws;
  float* dinv = (float*)ws;  ws += align256((size_t)N * 4);
  float* cnt  = (float*)ws;  ws += align256((size_t)G * 4);
  float* bufA = (float*)ws;  ws += align256((size_t)N * 256 * 4);
  float* bufB = (float*)ws;  ws += align256((size_t)N * 256 * 4);

  float* pool = (float*)d_out;          // [G, H2]
  float* hout = (float*)d_out + (size_t)G * H2;  // [N, H2]

  const int B = 256;
  auto cdiv = [](long long a, long long b) { return (int)((a + b - 1) / b); };

  // deg -> dinv (shared by all layers: dst set is fixed)
  k_fill<<<cdiv(N, B), B, 0, stream>>>(dinv, 1.0f, N);
  k_deg_edges<<<cdiv(E, B), B, 0, stream>>>(dstp, dinv, E);
  k_dinv<<<cdiv(N, B), B, 0, stream>>>(dinv, N);

  auto layer = [&](const float* hin, int K, int M, const float* W, const float* bias,
                   float* hw, float* agg, float* outp) {
    dim3 g(cdiv(N, 16), M / 128);
    k_gemm_wmma<<<g, B, 0, stream>>>(hin, W, hw, N, K, M);
    k_selfloop<<<cdiv((long long)N * M, B), B, 0, stream>>>(hw, dinv, agg, N, M);
    k_edge_scatter<<<cdiv((long long)E * 32, B), B, 0, stream>>>(srcp, dstp, dinv, hw, agg, E, M);
    k_post<<<cdiv((long long)N * 32, B), B, 0, stream>>>(agg, bias, outp, N, M);
  };

  // Layer 0: x[N,IN] -> bufA (h1, [N,H0]);  hw=bufA is dead after scatter input read,
  // so ping-pong: gemm->bufA, agg->bufB, post->bufA
  layer(x,    IN, H0, W0, b0, /*hw=*/bufA, /*agg=*/bufB, /*out=*/bufA);
  // Layer 1: bufA -> bufB (h2, [N,H1]); gemm->bufB, agg->bufA, post->bufB
  layer(bufA, H0, H1, W1, b1, /*hw=*/bufB, /*agg=*/bufA, /*out=*/bufB);
  // Layer 2: bufB -> hout (final h, [N,H2]); gemm->bufA, agg->bufB, post->hout
  layer(bufB, H1, H2, W2, b2, /*hw=*/bufA, /*agg=*/bufB, /*out=*/hout);

  // global_mean_pool into d_out[0 : G*H2]
  k_fill<<<cdiv((long long)G * H2, B), B, 0, stream>>>(pool, 0.0f, G * H2);
  k_fill<<<cdiv(G, B), B, 0, stream>>>(cnt, 0.0f, G);
  k_pool_accum<<<cdiv((long long)N * H2, B), B, 0, stream>>>(hout, batch, pool, N, H2);
  k_count<<<cdiv(N, B), B, 0, stream>>>(batch, cnt, N);
  k_pool_final<<<cdiv((long long)G * H2, B), B, 0, stream>>>(pool, cnt, G, H2);
}